// STDP_32521492366031
// MI455X (gfx1250) — compile-verified
//
#include <hip/hip_runtime.h>
#include <hip/hip_bf16.h>

// Problem constants (from reference)
#define T_STEPS 15
#define CIN     32
#define H_IN    256
#define W_IN    256
#define F_MAPS  64
#define KH      5
#define KW      5
#define CP      (CIN * KH * KW)   // 800 = contribution elements per winner
#define NTILE   (CP / 16)         // 50 WMMA M-tiles of 16 rows
#define DECAY   0.95f

typedef __attribute__((ext_vector_type(2))) float v2f;
typedef __attribute__((ext_vector_type(8))) float v8f;

// Kernel A: out[i] = clip(weight[i], 0, 1) for every feature map.
// Winner maps are fully overwritten by kernel B afterwards (stream-ordered).
__global__ __launch_bounds__(256) void stdp_clip_copy(const float* __restrict__ w,
                                                      float* __restrict__ out, int n) {
    int i = blockIdx.x * blockDim.x + threadIdx.x;
    if (i < n) {
        float v = w[i];
        out[i] = fminf(fmaxf(v, 0.0f), 1.0f);
    }
}

// Kernel B: one block (8 wave32) per winner.
// Phase 1: contribution[c*25+p] = sum_t fac[t] * rf[t,c,p] via V_WMMA_F32_16X16X4_F32
//   A (16x4) = RF rows (16 of the 800 (c,p) rows) x 4 timesteps, loaded from global
//   B (4x16) = decay factors for the 4 timesteps, identical in every column
//   D column 0 = 16 contribution values (lanes 0 and 16 hold it) -> LDS
// Phase 2: per-position (25) channel argmax + sum mask.
// Phase 3: rewrite feature map f with clipped stabilized update.
__global__ __launch_bounds__(256) void stdp_winner_kernel(
    const float* __restrict__ spikes,   // [T, CIN, H, W]
    const float* __restrict__ weight,   // [F, CIN, KH, KW]
    const float* __restrict__ ltp,      // [F]
    const float* __restrict__ ltd,      // [F]
    const int*   __restrict__ winners,  // [K, 4] = (t_out, f, h, w)
    float* __restrict__ out)            // [F, CIN, KH, KW]
{
    __shared__ float facs[16];     // decay factors, t=15 padded to 0
    __shared__ float contrib[CP];  // [CIN*25]
    __shared__ int   winch[KH * KW];
    __shared__ float psum[KH * KW];

    const int k     = blockIdx.x;
    const int t_out = winners[k * 4 + 0];
    const int f     = winners[k * 4 + 1];
    const int h0    = winners[k * 4 + 2];
    const int w0    = winners[k * 4 + 3];

    const int tid  = threadIdx.x;
    const int lane = tid & 31;
    const int wv   = tid >> 5;

    // factors: fac[t] = DECAY^(t_out - t) for t <= t_out else 0; fac[15] = 0
    if (tid < 16) {
        float v = 0.0f;
        if (tid <= t_out) {
            float p = 1.0f;
            for (int i = 0; i < t_out - tid; ++i) p *= DECAY;
            v = p;
        }
        facs[tid] = v;
    }
    __syncthreads();

    // ---- Phase 1: WMMA time-contraction ----------------------------------
    const int tplane = CIN * H_IN * W_IN;  // stride between timesteps
    for (int tile = wv; tile < NTILE; tile += 8) {   // uniform per wave -> EXEC all 1s at WMMA
        const int m  = tile * 16 + (lane & 15);      // (c,p) row index, 0..799
        const int c  = m / 25;
        const int p  = m % 25;
        const int kh = p / 5;
        const int kw = p % 5;
        const float* base = spikes + ((size_t)c * H_IN + (h0 + kh)) * W_IN + (w0 + kw);

        v8f acc = {};
        #pragma unroll
        for (int tc = 0; tc < 16; tc += 4) {
            const int t0 = tc + ((lane < 16) ? 0 : 2);
            const int t1 = t0 + 1;
            v2f a, b;
            a.x = (t0 < T_STEPS) ? base[(size_t)t0 * tplane] : 0.0f;
            a.y = (t1 < T_STEPS) ? base[(size_t)t1 * tplane] : 0.0f;
            b.x = facs[t0];
            b.y = facs[t1];
            // (neg_a, A, neg_b, B, c_mod, C, reuse_a, reuse_b)
            acc = __builtin_amdgcn_wmma_f32_16x16x4_f32(
                      false, a, false, b, (short)0, acc, false, false);
        }
        // Column 0 of D: lane 0 holds rows m0+0..7 (vgpr 0..7), lane 16 rows m0+8..15
        if (lane == 0 || lane == 16) {
            const int r0 = tile * 16 + ((lane == 16) ? 8 : 0);
            #pragma unroll
            for (int i = 0; i < 8; ++i) contrib[r0 + i] = acc[i];
        }
    }
    __syncthreads();

    // ---- Phase 2: per-position argmax over channels + spike mask sum -----
    if (tid < KH * KW) {
        float best = contrib[tid];   // c = 0
        float s    = best;
        int   bc   = 0;
        for (int c = 1; c < CIN; ++c) {
            float v = contrib[c * 25 + tid];
            s += v;
            if (v > best) { best = v; bc = c; }  // strict > == jnp first-max
        }
        winch[tid] = bc;
        psum[tid]  = s;
    }
    __syncthreads();

    // ---- Phase 3: stabilized STDP update of feature map f ----------------
    const float ltpf = ltp[f];
    const float ltdf = ltd[f];
    for (int i = tid; i < CP; i += 256) {
        const int c = i / 25;
        const int p = i % 25;
        const float wgt  = weight[f * CP + i];
        const float stab = wgt * (1.0f - wgt);          // (w-LB)*(UB-w)
        const bool isltp = (psum[p] > 0.0f) && (c == winch[p]);
        const float upd  = (isltp ? ltpf : ltdf) * stab;
        float o = wgt + upd;
        o = fminf(fmaxf(o, 0.0f), 1.0f);
        out[f * CP + i] = o;
    }
}

extern "C" void kernel_launch(void* const* d_in, const int* in_sizes, int n_in,
                              void* d_out, int out_size, void* d_ws, size_t ws_size,
                              hipStream_t stream) {
    const float* input_spikes = (const float*)d_in[0];  // [T,CIN,H,W]
    // d_in[1] potentials, d_in[2] output_spikes: unused by the math
    const float* weight  = (const float*)d_in[3];       // [F,CIN,KH,KW]
    const float* ltp     = (const float*)d_in[4];       // [F]
    const float* ltd     = (const float*)d_in[5];       // [F]
    const int*   winners = (const int*)d_in[6];         // [K,4]
    float*       out     = (float*)d_out;               // [F,CIN,KH,KW]

    const int n = F_MAPS * CP;  // 51200
    stdp_clip_copy<<<(n + 255) / 256, 256, 0, stream>>>(weight, out, n);
    stdp_winner_kernel<<<64, 256, 0, stream>>>(input_spikes, weight, ltp, ltd,
                                               winners, out);
}